// LocalPPFTransformer_53214644797678
// MI455X (gfx1250) — compile-verified
//
#include <hip/hip_runtime.h>
#include <hip/hip_bf16.h>

// ---------------------------------------------------------------------------
// LocalPPFTransformer for MI455X (gfx1250, wave32).
// Sizes: N=50000, M=20000, K=32, IN=64, D=128, H=8, DH=16, OUT=128.
// Positional MLP folded algebraically: p = ppf @ (W_embed@Wp) + (b_embed@Wp+bp)
// ---------------------------------------------------------------------------

typedef __attribute__((ext_vector_type(2))) float v2f;
typedef __attribute__((ext_vector_type(8))) float v8f;

#define D_DIM   128
#define IN_DIM  64
#define KGRP    32
#define LDS_STR 132   // padded row stride (floats) for 16xD tiles
#define EPS_LN  1e-5f

__device__ __forceinline__ v8f wmma4(v2f a, v2f b, v8f c) {
  // v_wmma_f32_16x16x4_f32 : D = A(16x4) * B(4x16) + C(16x16), fp32
  return __builtin_amdgcn_wmma_f32_16x16x4_f32(
      /*neg_a=*/false, a, /*neg_b=*/false, b,
      /*c_mod=*/(short)0, c, /*reuse_a=*/false, /*reuse_b=*/false);
}

// 16x16 output tile: A (16xK) read from LDS (row stride lda), B (Kx128
// row-major, ld=128) read from global. half = lane>>4, r = lane&15.
__device__ __forceinline__ v8f gemm_tile(const float* sA, int lda,
                                         const float* __restrict__ W,
                                         int K, int n0, int half, int r) {
  v8f c = {};
  for (int k0 = 0; k0 < K; k0 += 4) {
    v2f a, b;
    const float* ap = sA + r * lda + k0 + 2 * half;
    a.x = ap[0];
    a.y = ap[1];
    const float* bp = W + (size_t)(k0 + 2 * half) * D_DIM + n0 + r;
    b.x = bp[0];
    b.y = bp[D_DIM];
    c = wmma4(a, b, c);
  }
  return c;
}

// ---------------------------------------------------------------------------
// K0: fold the positional embedding into effective 4x128 weights + biases.
// ---------------------------------------------------------------------------
__global__ __launch_bounds__(128) void k0_fold(
    const float* __restrict__ W_embed, const float* __restrict__ b_embed,
    const float* __restrict__ Wp, const float* __restrict__ bp,
    const float* __restrict__ Wvp, const float* __restrict__ bvp,
    float* __restrict__ wp_eff, float* __restrict__ bp_eff,
    float* __restrict__ wvp_eff, float* __restrict__ bvp_eff) {
  int j = threadIdx.x;  // output column 0..127
  float ap[4] = {0.f, 0.f, 0.f, 0.f};
  float av[4] = {0.f, 0.f, 0.f, 0.f};
  float bpj = bp[j], bvj = bvp[j];
  for (int dd = 0; dd < D_DIM; ++dd) {
    float wpj = Wp[(size_t)dd * D_DIM + j];
    float wvj = Wvp[(size_t)dd * D_DIM + j];
    float be = b_embed[dd];
    bpj += be * wpj;
    bvj += be * wvj;
#pragma unroll
    for (int i = 0; i < 4; ++i) {
      float we = W_embed[i * D_DIM + dd];
      ap[i] += we * wpj;
      av[i] += we * wvj;
    }
  }
#pragma unroll
  for (int i = 0; i < 4; ++i) {
    wp_eff[i * D_DIM + j] = ap[i];
    wvp_eff[i * D_DIM + j] = av[i];
  }
  bp_eff[j] = bpj;
  bvp_eff[j] = bvj;
}

// ---------------------------------------------------------------------------
// K1: h = feats@W_in+b_in; hq/hk/hv = h@Wq/Wk/Wv + b. One wave per 16 rows.
// ---------------------------------------------------------------------------
__global__ __launch_bounds__(32) void k1_proj(
    const float* __restrict__ feats,
    const float* __restrict__ W_in, const float* __restrict__ b_in,
    const float* __restrict__ Wq, const float* __restrict__ bq,
    const float* __restrict__ Wk, const float* __restrict__ bk,
    const float* __restrict__ Wv, const float* __restrict__ bv,
    float* __restrict__ h, float* __restrict__ hq,
    float* __restrict__ hk, float* __restrict__ hv) {
  __shared__ float sF[16 * 68];        // feats tile (16 x 64, padded)
  __shared__ float sH[16 * LDS_STR];   // h tile (16 x 128, padded)

  const int tid = threadIdx.x;
  const int half = tid >> 4, r = tid & 15;
  const int m0 = blockIdx.x * 16;

  // stage feats tile (16x64) into LDS, float4 granularity
  for (int t = tid; t < 16 * 16; t += 32) {
    int row = t >> 4, c4 = (t & 15) * 4;
    *(float4*)&sF[row * 68 + c4] =
        *(const float4*)(feats + (size_t)(m0 + row) * IN_DIM + c4);
  }
  __syncthreads();

  // h = feats @ W_in + b_in  (K = 64)
  for (int t = 0; t < 8; ++t) {
    int n0 = t * 16;
    v8f c = gemm_tile(sF, 68, W_in, IN_DIM, n0, half, r);
    float bias = b_in[n0 + r];
#pragma unroll
    for (int i = 0; i < 8; ++i) {
      int row = i + 8 * half;
      float val = c[i] + bias;
      h[(size_t)(m0 + row) * D_DIM + n0 + r] = val;
      sH[row * LDS_STR + n0 + r] = val;
    }
  }
  __syncthreads();

  // hq/hk/hv = h @ W + b  (K = 128), reusing the LDS h tile
  const float* Ws[3] = {Wq, Wk, Wv};
  const float* bs[3] = {bq, bk, bv};
  float* outs[3] = {hq, hk, hv};
  for (int p = 0; p < 3; ++p) {
    for (int t = 0; t < 8; ++t) {
      int n0 = t * 16;
      v8f c = gemm_tile(sH, LDS_STR, Ws[p], D_DIM, n0, half, r);
      float bias = bs[p][n0 + r];
#pragma unroll
      for (int i = 0; i < 8; ++i) {
        int row = i + 8 * half;
        outs[p][(size_t)(m0 + row) * D_DIM + n0 + r] = c[i] + bias;
      }
    }
  }
}

// ---------------------------------------------------------------------------
// K2: per-m attention. One wave per m; lane = group slot k (K=32 = wave32).
// ---------------------------------------------------------------------------
__global__ __launch_bounds__(32) void k2_attn(
    const float* __restrict__ hq, const float* __restrict__ hk,
    const float* __restrict__ hv,
    const int* __restrict__ node_idx, const int* __restrict__ group_idx,
    const float* __restrict__ ppfs,
    const float* __restrict__ wp_eff, const float* __restrict__ bp_eff,
    const float* __restrict__ wvp_eff, const float* __restrict__ bvp_eff,
    float* __restrict__ hidden) {
  __shared__ float wbuf[KGRP * LDS_STR];

  const int m = blockIdx.x;
  const int k = threadIdx.x;  // 0..31

  const int nidx = node_idx[m];
  const int gidx = group_idx[m * KGRP + k];
  const float4 pf = *(const float4*)(ppfs + (size_t)(m * KGRP + k) * 4);

  const float* qrow = hq + (size_t)nidx * D_DIM;  // uniform across lanes
  const float* krow = hk + (size_t)gidx * D_DIM;
  const float* vrow = hv + (size_t)gidx * D_DIM;

  // --- phase 1: per-head scores s[h] = q . (k + p) -------------------------
  float s[8] = {0.f, 0.f, 0.f, 0.f, 0.f, 0.f, 0.f, 0.f};
  for (int d0 = 0; d0 < D_DIM; d0 += 4) {
    float4 kv = *(const float4*)(krow + d0);
    float4 q4 = *(const float4*)(qrow + d0);
    int hh = d0 >> 4;
    float kvj[4] = {kv.x, kv.y, kv.z, kv.w};
    float qj[4] = {q4.x, q4.y, q4.z, q4.w};
    float pfc[4] = {pf.x, pf.y, pf.z, pf.w};
#pragma unroll
    for (int j = 0; j < 4; ++j) {
      int d = d0 + j;
      float pj = bp_eff[d];
#pragma unroll
      for (int i = 0; i < 4; ++i) pj += pfc[i] * wp_eff[i * D_DIM + d];
      s[hh] += qj[j] * (kvj[j] + pj);
    }
  }

  // --- softmax over k (across lanes) per head ------------------------------
  float attn[8];
#pragma unroll
  for (int hh = 0; hh < 8; ++hh) {
    float sc = s[hh] * 0.25f;  // 1/sqrt(16)
    float mx = sc;
#pragma unroll
    for (int off = 16; off >= 1; off >>= 1)
      mx = fmaxf(mx, __shfl_xor(mx, off, 32));
    float e = __expf(sc - mx);
    float sm = e;
#pragma unroll
    for (int off = 16; off >= 1; off >>= 1) sm += __shfl_xor(sm, off, 32);
    attn[hh] = e / sm;
  }

  // --- phase 2: weighted values into LDS, reduce over k --------------------
  for (int d0 = 0; d0 < D_DIM; d0 += 4) {
    float4 vv = *(const float4*)(vrow + d0);
    int hh = d0 >> 4;
    float vvj[4] = {vv.x, vv.y, vv.z, vv.w};
    float pfc[4] = {pf.x, pf.y, pf.z, pf.w};
    float4 w;
    float wj[4];
#pragma unroll
    for (int j = 0; j < 4; ++j) {
      int d = d0 + j;
      float pj = bvp_eff[d];
#pragma unroll
      for (int i = 0; i < 4; ++i) pj += pfc[i] * wvp_eff[i * D_DIM + d];
      wj[j] = attn[hh] * (vvj[j] + pj);
    }
    w.x = wj[0]; w.y = wj[1]; w.z = wj[2]; w.w = wj[3];
    *(float4*)&wbuf[k * LDS_STR + d0] = w;
  }
  __syncthreads();

#pragma unroll
  for (int t = 0; t < 4; ++t) {
    int d = k + 32 * t;
    float acc = 0.f;
    for (int kk = 0; kk < KGRP; ++kk) acc += wbuf[kk * LDS_STR + d];
    hidden[(size_t)m * D_DIM + d] = acc;
  }
}

// ---------------------------------------------------------------------------
// K3: x = hidden@Wl + bl + h[node_idx]; LayerNorm; out = xn@Wout + bout.
// One wave per 16 rows of M; both GEMMs via WMMA.
// ---------------------------------------------------------------------------
__global__ __launch_bounds__(32) void k3_out(
    const float* __restrict__ hidden, const float* __restrict__ h,
    const int* __restrict__ node_idx,
    const float* __restrict__ Wl, const float* __restrict__ bl,
    const float* __restrict__ gamma, const float* __restrict__ beta,
    const float* __restrict__ Wout, const float* __restrict__ bout,
    float* __restrict__ out) {
  __shared__ float sA[16 * LDS_STR];  // hidden tile, later xn tile
  __shared__ float sB[16 * LDS_STR];  // x tile
  __shared__ int sIdx[16];

  const int tid = threadIdx.x;
  const int half = tid >> 4, r = tid & 15;
  const int m0 = blockIdx.x * 16;

  for (int t = tid; t < 16 * 32; t += 32) {
    int row = t >> 5, c4 = (t & 31) * 4;
    *(float4*)&sA[row * LDS_STR + c4] =
        *(const float4*)(hidden + (size_t)(m0 + row) * D_DIM + c4);
  }
  if (tid < 16) sIdx[tid] = node_idx[m0 + tid];
  __syncthreads();

  // x = hidden @ Wl + bl + h[node_idx]
  for (int t = 0; t < 8; ++t) {
    int n0 = t * 16;
    v8f c = gemm_tile(sA, LDS_STR, Wl, D_DIM, n0, half, r);
    float bias = bl[n0 + r];
#pragma unroll
    for (int i = 0; i < 8; ++i) {
      int row = i + 8 * half;
      float x = c[i] + bias + h[(size_t)sIdx[row] * D_DIM + n0 + r];
      sB[row * LDS_STR + n0 + r] = x;
    }
  }
  __syncthreads();

  // LayerNorm each row (16 lanes, one row each)
  if (tid < 16) {
    int row = tid;
    float sum = 0.f;
    for (int c = 0; c < D_DIM; ++c) sum += sB[row * LDS_STR + c];
    float mu = sum * (1.0f / D_DIM);
    float var = 0.f;
    for (int c = 0; c < D_DIM; ++c) {
      float d = sB[row * LDS_STR + c] - mu;
      var += d * d;
    }
    var *= (1.0f / D_DIM);
    float rs = rsqrtf(var + EPS_LN);
    for (int c = 0; c < D_DIM; ++c) {
      float xn = (sB[row * LDS_STR + c] - mu) * rs;
      sA[row * LDS_STR + c] = xn * gamma[c] + beta[c];
    }
  }
  __syncthreads();

  // out = xn @ Wout + bout
  for (int t = 0; t < 8; ++t) {
    int n0 = t * 16;
    v8f c = gemm_tile(sA, LDS_STR, Wout, D_DIM, n0, half, r);
    float bias = bout[n0 + r];
#pragma unroll
    for (int i = 0; i < 8; ++i) {
      int row = i + 8 * half;
      out[(size_t)(m0 + row) * D_DIM + n0 + r] = c[i] + bias;
    }
  }
}

// ---------------------------------------------------------------------------
extern "C" void kernel_launch(void* const* d_in, const int* in_sizes, int n_in,
                              void* d_out, int out_size, void* d_ws,
                              size_t ws_size, hipStream_t stream) {
  const float* feats    = (const float*)d_in[0];
  const int*   node_idx = (const int*)d_in[1];
  const int*   group_idx= (const int*)d_in[2];
  const float* ppfs     = (const float*)d_in[3];
  const float* W_embed  = (const float*)d_in[4];
  const float* b_embed  = (const float*)d_in[5];
  const float* W_in     = (const float*)d_in[6];
  const float* b_in     = (const float*)d_in[7];
  const float* Wq       = (const float*)d_in[8];
  const float* bq       = (const float*)d_in[9];
  const float* Wk       = (const float*)d_in[10];
  const float* bk       = (const float*)d_in[11];
  const float* Wv       = (const float*)d_in[12];
  const float* bv       = (const float*)d_in[13];
  const float* Wp       = (const float*)d_in[14];
  const float* bp       = (const float*)d_in[15];
  const float* Wvp      = (const float*)d_in[16];
  const float* bvp      = (const float*)d_in[17];
  const float* Wl       = (const float*)d_in[18];
  const float* bl       = (const float*)d_in[19];
  const float* gamma    = (const float*)d_in[20];
  const float* beta     = (const float*)d_in[21];
  const float* Wout     = (const float*)d_in[22];
  const float* bout     = (const float*)d_in[23];

  const int N = in_sizes[0] / IN_DIM;  // 50000
  const int M = in_sizes[1];           // 20000

  // workspace layout (floats)
  float* ws = (float*)d_ws;
  float* h      = ws;                     // N*128
  float* hq     = h      + (size_t)N * D_DIM;
  float* hk     = hq     + (size_t)N * D_DIM;
  float* hv     = hk     + (size_t)N * D_DIM;
  float* hidden = hv     + (size_t)N * D_DIM;   // M*128
  float* wp_eff = hidden + (size_t)M * D_DIM;   // 512
  float* bp_eff = wp_eff + 4 * D_DIM;           // 128
  float* wvp_eff= bp_eff + D_DIM;               // 512
  float* bvp_eff= wvp_eff+ 4 * D_DIM;           // 128

  k0_fold<<<1, 128, 0, stream>>>(W_embed, b_embed, Wp, bp, Wvp, bvp,
                                 wp_eff, bp_eff, wvp_eff, bvp_eff);

  k1_proj<<<N / 16, 32, 0, stream>>>(feats, W_in, b_in, Wq, bq, Wk, bk, Wv, bv,
                                     h, hq, hk, hv);

  k2_attn<<<M, 32, 0, stream>>>(hq, hk, hv, node_idx, group_idx, ppfs,
                                wp_eff, bp_eff, wvp_eff, bvp_eff, hidden);

  k3_out<<<M / 16, 32, 0, stream>>>(hidden, h, node_idx, Wl, bl, gamma, beta,
                                    Wout, bout, (float*)d_out);
}